// Network_22892175688023
// MI455X (gfx1250) — compile-verified
//
#include <hip/hip_runtime.h>
#include <math.h>

// ---------------------------------------------------------------------------
// Problem constants (match reference)
// ---------------------------------------------------------------------------
#define B_    1024
#define DIN   103
#define DINP  104     // DIN padded to multiple of 4 for WMMA K-loop
#define L1_   64
#define L2_   32
#define E_    8
#define GH_   64
#define H_    256
#define DOUT_ 51
#define KG1   (E_*DIN)   // 824  (multiple of 4)
#define KG2   (E_*H_)    // 2048 (multiple of 4)
#define EPSB  1e-5f

typedef __attribute__((ext_vector_type(2))) float v2f;
typedef __attribute__((ext_vector_type(4))) float v4f;
typedef __attribute__((ext_vector_type(8))) float v8f;

// CDNA5 async global->LDS path (ASYNCcnt-tracked), if the toolchain has it.
#if defined(__has_builtin)
#if __has_builtin(__builtin_amdgcn_global_load_async_to_lds_b128)
#define HAVE_ASYNC_LDS 1
#endif
#endif
#ifndef HAVE_ASYNC_LDS
#define HAVE_ASYNC_LDS 0
#endif

#if HAVE_ASYNC_LDS
// builtin expects: (v4i addrspace(1)*, v4i addrspace(3)*, imm int, imm int)
typedef __attribute__((ext_vector_type(4))) int i4v;
typedef __attribute__((address_space(1))) i4v gi4v;   // global (AS1)
typedef __attribute__((address_space(3))) i4v li4v;   // LDS (AS3)
__device__ __forceinline__ void async_copy16(const float* src, float* dst) {
  __builtin_amdgcn_global_load_async_to_lds_b128(
      (gi4v*)(i4v*)(void*)src, (li4v*)(i4v*)(void*)dst, 0, 0);
}
#endif

// ---------------------------------------------------------------------------
// Elementwise prep: scale last 3 features by 100, pad K 103 -> 104 with zeros
// ---------------------------------------------------------------------------
__global__ void prep_x(const float* __restrict__ x, float* __restrict__ xs) {
  int idx = blockIdx.x * 256 + threadIdx.x;
  if (idx >= B_ * DINP) return;
  int b = idx / DINP, i = idx - b * DINP;
  float v = 0.0f;
  if (i < DIN) {
    v = x[b * DIN + i];
    if (i >= 100) v *= 100.0f;
  }
  xs[idx] = v;
}

// w1 is [L1, DIN] row-major; build padded [DINP, L1] (K-major) with zero rows
__global__ void prep_w1t(const float* __restrict__ w1, float* __restrict__ w1t) {
  int idx = blockIdx.x * 256 + threadIdx.x;
  if (idx >= DINP * L1_) return;
  int k = idx / L1_, n = idx - k * L1_;
  w1t[idx] = (k < DIN) ? w1[n * DIN + k] : 0.0f;
}

// ---------------------------------------------------------------------------
// Generic register-only fp32 WMMA GEMM (small K: encoder + gating layers)
//   BT=false: W stored [K,N] row-major;  BT=true: W stored [N,K] row-major
//   BIAS: 0=none, 1=per-column bias[n];  ACT: 0=none, 1=ELU
// Block = 128 threads = 4 waves; wave w owns N-tile (blockIdx.x*4 + w).
// ---------------------------------------------------------------------------
template <int BIAS, int ACT, bool BT>
__global__ void wmma_gemm(const float* __restrict__ A, int lda,
                          const float* __restrict__ W, int ldw,
                          const float* __restrict__ bias,
                          float* __restrict__ C, int ldc,
                          int M, int N, int K) {
  const int lane = threadIdx.x & 31;
  const int wave = threadIdx.x >> 5;
  const int lo = lane & 15;
  const int g  = lane >> 4;            // lane-half: selects K pair
  const int m0 = blockIdx.y * 16;
  const int n0 = (blockIdx.x * 4 + wave) * 16;
  if (n0 >= N) return;                 // uniform per wave: no WMMA issued

  const int row  = m0 + lo;
  const int col  = n0 + lo;
  const int colC = (col < N) ? col : (N - 1);   // clamp to stay in-bounds

  const float* aPtr = A + (size_t)row * lda + 2 * g;
  const float* bPtr = BT ? (W + (size_t)colC * ldw + 2 * g)
                         : (W + (size_t)(2 * g) * ldw + colC);

  v8f acc = {0.f, 0.f, 0.f, 0.f, 0.f, 0.f, 0.f, 0.f};

  for (int k = 0; k < K; k += 4) {
    v2f a = *(const v2f*)(aPtr + k);         // A[row][k+2g], A[row][k+2g+1]
    v2f b;
    if (BT) {
      b = *(const v2f*)(bPtr + k);           // W[colC][k+2g], W[colC][k+2g+1]
    } else {
      b.x = bPtr[(size_t)k * ldw];           // W[k+2g  ][colC]
      b.y = bPtr[(size_t)k * ldw + ldw];     // W[k+2g+1][colC]
    }
    acc = __builtin_amdgcn_wmma_f32_16x16x4_f32(
        false, a, false, b, (short)0, acc, false, false);
  }

#pragma unroll
  for (int v = 0; v < 8; ++v) {
    int r = m0 + v + 8 * g;
    int c = n0 + lo;
    if (c < N) {
      float val = acc[v];
      if (BIAS == 1) val += bias[c];
      if (ACT == 1) val = (val > 0.0f) ? val : (__expf(val) - 1.0f);  // ELU
      C[(size_t)r * ldc + c] = val;
    }
  }
}

// ---------------------------------------------------------------------------
// Generator WMMA GEMM with LDS-staged A (large K, W stored [K,N]).
//   C[M,N] = act(A[M,K] @ W + rowBias[M,N])
// Block = 128 threads = 4 waves, computes a 64(M) x 64(N) C block.
//   - A block tile (64 rows x KC cols) staged into LDS, double-buffered,
//     via async global->LDS copies (ASYNCcnt) when available.
//   - Wave w owns N-subtile w: 4 accumulators (64M x 16N) share one B
//     fragment per K-step -> 4 WMMA per B load pair.
// LDS row stride 68 floats: 272B rows (16B-aligned for b128 async writes)
// and a 4-bank skew so b64 fragment reads across 16 lanes hit 16 banks.
// ---------------------------------------------------------------------------
#define KC   64
#define LDSS 68

template <int ACT>
__global__ void wmma_gemm_gen(const float* __restrict__ A, int lda,
                              const float* __restrict__ W, int ldw,
                              const float* __restrict__ rb, int ldrb,
                              float* __restrict__ C, int ldc,
                              int M, int N, int K) {
  __shared__ float As[2][64 * LDSS];

  const int lane = threadIdx.x & 31;
  const int wave = threadIdx.x >> 5;
  const int lo = lane & 15;
  const int g  = lane >> 4;
  const int m0 = blockIdx.y * 64;
  const int n0 = blockIdx.x * 64 + wave * 16;   // always < N for our shapes

  const int col  = n0 + lo;
  const int colC = (col < N) ? col : (N - 1);
  const float* bPtr = W + (size_t)(2 * g) * ldw + colC;

  v8f acc[4];
#pragma unroll
  for (int s = 0; s < 4; ++s)
    acc[s] = (v8f){0.f, 0.f, 0.f, 0.f, 0.f, 0.f, 0.f, 0.f};

  // stage chunk [k0, k0+kc) of the 64-row A block into LDS buffer `buf`
  auto stage = [&](int buf, int k0, int kc) {
    const int quads = kc >> 2;            // 16-byte quads per row (<= 16)
    const int total = 64 * quads;
    for (int q = threadIdx.x; q < total; q += 128) {
      int r = q / quads, qq = q - r * quads;
      const float* src = A + (size_t)(m0 + r) * lda + k0 + qq * 4;
      float* dst = &As[buf][r * LDSS + qq * 4];
#if HAVE_ASYNC_LDS
      async_copy16(src, dst);
#else
      *(v4f*)dst = *(const v4f*)src;
#endif
    }
  };

  const int nCh = (K + KC - 1) / KC;

#if HAVE_ASYNC_LDS
  stage(0, 0, KC < K ? KC : K);                       // prologue
  for (int c = 0; c < nCh; ++c) {
    const int k0 = c * KC;
    const int kc = (K - k0) < KC ? (K - k0) : KC;
    const int buf = c & 1;
    asm volatile("s_wait_asynccnt 0x0" ::: "memory"); // this wave's stage done
    __syncthreads();                                  // all waves' stages done
    if (c + 1 < nCh) {                                // prefetch next chunk
      const int kn0 = k0 + KC;
      const int knc = (K - kn0) < KC ? (K - kn0) : KC;
      stage(buf ^ 1, kn0, knc);
    }
#else
  for (int c = 0; c < nCh; ++c) {
    const int k0 = c * KC;
    const int kc = (K - k0) < KC ? (K - k0) : KC;
    const int buf = 0;
    __syncthreads();                                  // prior reads done
    stage(buf, k0, kc);
    __syncthreads();                                  // staged data visible
#endif
#pragma unroll 4
    for (int kk = 0; kk < kc; kk += 4) {
      v2f b;
      b.x = bPtr[(size_t)(k0 + kk) * ldw];            // W[k+2g  ][colC]
      b.y = bPtr[(size_t)(k0 + kk) * ldw + ldw];      // W[k+2g+1][colC]
#pragma unroll
      for (int s = 0; s < 4; ++s) {
        v2f a = *(const v2f*)&As[buf][(s * 16 + lo) * LDSS + kk + 2 * g];
        acc[s] = __builtin_amdgcn_wmma_f32_16x16x4_f32(
            false, a, false, b, (short)0, acc[s], false, false);
      }
    }
  }

#pragma unroll
  for (int s = 0; s < 4; ++s) {
#pragma unroll
    for (int v = 0; v < 8; ++v) {
      int r = m0 + s * 16 + v + 8 * g;
      int c = n0 + lo;
      if (c < N) {
        float val = acc[s][v] + rb[(size_t)r * ldrb + c];
        if (ACT == 1) val = (val > 0.0f) ? val : (__expf(val) - 1.0f);  // ELU
        C[(size_t)r * ldc + c] = val;
      }
    }
  }
}

// ---------------------------------------------------------------------------
// BatchNorm (training mode): per-column mean / biased var over B_ rows.
// ---------------------------------------------------------------------------
__global__ void bn_stats(const float* __restrict__ h, int N,
                         const float* __restrict__ gamma,
                         const float* __restrict__ beta,
                         float* __restrict__ scale, float* __restrict__ shift) {
  __shared__ float s_sum[256];
  __shared__ float s_sq[256];
  int c = blockIdx.x;
  float s = 0.f, q = 0.f;
  for (int r = threadIdx.x; r < B_; r += 256) {
    float v = h[(size_t)r * N + c];
    s += v;
    q += v * v;
  }
  s_sum[threadIdx.x] = s;
  s_sq[threadIdx.x] = q;
  __syncthreads();
  for (int off = 128; off > 0; off >>= 1) {
    if (threadIdx.x < off) {
      s_sum[threadIdx.x] += s_sum[threadIdx.x + off];
      s_sq[threadIdx.x]  += s_sq[threadIdx.x + off];
    }
    __syncthreads();
  }
  if (threadIdx.x == 0) {
    float mean = s_sum[0] * (1.0f / B_);
    float var  = s_sq[0] * (1.0f / B_) - mean * mean;   // biased var
    float sc   = gamma[c] * rsqrtf(var + EPSB);
    scale[c] = sc;
    shift[c] = beta[c] - mean * sc;
  }
}

__global__ void bn_apply_relu(float* __restrict__ h, int N,
                              const float* __restrict__ scale,
                              const float* __restrict__ shift) {
  int idx = blockIdx.x * 256 + threadIdx.x;
  if (idx >= B_ * N) return;
  int c = idx % N;   // N is 64 or 32
  float v = h[idx] * scale[c] + shift[c];
  h[idx] = v > 0.0f ? v : 0.0f;
}

// ---------------------------------------------------------------------------
// Softmax over E_=8 gating logits, in place. One thread per batch row.
// ---------------------------------------------------------------------------
__global__ void softmax8(float* __restrict__ z) {
  int b = blockIdx.x * 256 + threadIdx.x;
  if (b >= B_) return;
  float* p = z + b * E_;
  float m = p[0];
#pragma unroll
  for (int e = 1; e < E_; ++e) m = fmaxf(m, p[e]);
  float v[E_], s = 0.f;
#pragma unroll
  for (int e = 0; e < E_; ++e) { v[e] = __expf(p[e] - m); s += v[e]; }
  float inv = 1.0f / s;
#pragma unroll
  for (int e = 0; e < E_; ++e) p[e] = v[e] * inv;
}

// ---------------------------------------------------------------------------
// A'-builders: fold the expert blend into the GEMM K dimension.
//   A1'[b, e*DIN+i] = bc[b,e]*xs[b,i]       (K = 824)
//   RB [b, n]       = sum_e bc[b,e]*eb[e,n] (per-row bias)
// ---------------------------------------------------------------------------
__global__ void build_a1(const float* __restrict__ xs, const float* __restrict__ bc,
                         const float* __restrict__ eb1,
                         float* __restrict__ A1, float* __restrict__ RB1) {
  int b = blockIdx.x;
  __shared__ float bcs[E_];
  if (threadIdx.x < E_) bcs[threadIdx.x] = bc[b * E_ + threadIdx.x];
  __syncthreads();
  for (int i = threadIdx.x; i < DIN; i += blockDim.x) {
    float xv = xs[(size_t)b * DINP + i];
#pragma unroll
    for (int e = 0; e < E_; ++e) A1[(size_t)b * KG1 + e * DIN + i] = bcs[e] * xv;
  }
  for (int n = threadIdx.x; n < H_; n += blockDim.x) {
    float s = 0.f;
#pragma unroll
    for (int e = 0; e < E_; ++e) s += bcs[e] * eb1[e * H_ + n];
    RB1[(size_t)b * H_ + n] = s;
  }
}

// A2'[b, e*H+h] = bc[b,e]*hin[b,h]  (K = 2048), plus row-bias of width ebN
__global__ void build_a2(const float* __restrict__ hin, const float* __restrict__ bc,
                         const float* __restrict__ eb, int ebN,
                         float* __restrict__ A2, float* __restrict__ RB) {
  int b = blockIdx.x;
  __shared__ float bcs[E_];
  if (threadIdx.x < E_) bcs[threadIdx.x] = bc[b * E_ + threadIdx.x];
  __syncthreads();
  for (int hcol = threadIdx.x; hcol < H_; hcol += blockDim.x) {
    float hv = hin[(size_t)b * H_ + hcol];
#pragma unroll
    for (int e = 0; e < E_; ++e) A2[(size_t)b * KG2 + e * H_ + hcol] = bcs[e] * hv;
  }
  for (int o = threadIdx.x; o < ebN; o += blockDim.x) {
    float s = 0.f;
#pragma unroll
    for (int e = 0; e < E_; ++e) s += bcs[e] * eb[e * ebN + o];
    RB[(size_t)b * ebN + o] = s;
  }
}

// ---------------------------------------------------------------------------
// Launcher
// ---------------------------------------------------------------------------
extern "C" void kernel_launch(void* const* d_in, const int* in_sizes, int n_in,
                              void* d_out, int out_size, void* d_ws, size_t ws_size,
                              hipStream_t stream) {
  (void)in_sizes; (void)n_in; (void)out_size; (void)ws_size;

  const float* x      = (const float*)d_in[0];
  const float* w1     = (const float*)d_in[1];
  const float* b1     = (const float*)d_in[2];
  const float* gamma1 = (const float*)d_in[3];
  const float* beta1  = (const float*)d_in[4];
  const float* w2     = (const float*)d_in[5];
  const float* b2     = (const float*)d_in[6];
  const float* gamma2 = (const float*)d_in[7];
  const float* beta2  = (const float*)d_in[8];
  const float* gw1    = (const float*)d_in[9];
  const float* gb1    = (const float*)d_in[10];
  const float* gw2    = (const float*)d_in[11];
  const float* gb2    = (const float*)d_in[12];
  const float* gw3    = (const float*)d_in[13];
  const float* gb3    = (const float*)d_in[14];
  const float* ew1    = (const float*)d_in[15];  // [E,DIN,H]  == [824,256]
  const float* eb1    = (const float*)d_in[16];
  const float* ew2    = (const float*)d_in[17];  // [E,H,H]    == [2048,256]
  const float* eb2    = (const float*)d_in[18];
  const float* ew3    = (const float*)d_in[19];  // [E,H,DOUT] == [2048,51]
  const float* eb3    = (const float*)d_in[20];
  float* out = (float*)d_out;

  // workspace carve-out (floats)
  float* w = (float*)d_ws;
  size_t off = 0;
  auto alloc = [&](size_t n) { float* p = w + off; off += n; return p; };
  float* xs  = alloc((size_t)B_ * DINP);
  float* w1t = alloc((size_t)DINP * L1_);
  float* hp1 = alloc((size_t)B_ * L1_);
  float* sc1 = alloc(L1_);
  float* sh1 = alloc(L1_);
  float* hp2 = alloc((size_t)B_ * L2_);
  float* sc2 = alloc(L2_);
  float* sh2 = alloc(L2_);
  float* g1  = alloc((size_t)B_ * GH_);
  float* g2  = alloc((size_t)B_ * GH_);
  float* bc  = alloc((size_t)B_ * E_);
  float* A1  = alloc((size_t)B_ * KG1);
  float* RB1 = alloc((size_t)B_ * H_);
  float* h1  = alloc((size_t)B_ * H_);
  float* A2  = alloc((size_t)B_ * KG2);   // reused for layer 3
  float* RB2 = alloc((size_t)B_ * H_);
  float* h2  = alloc((size_t)B_ * H_);
  float* RB3 = alloc((size_t)B_ * DOUT_);

  const int MT = B_ / 16;   // 64 16-row tiles
  const int MB = B_ / 64;   // 16 64-row blocks

  // ---- input prep ----
  prep_x<<<(B_ * DINP + 255) / 256, 256, 0, stream>>>(x, xs);
  prep_w1t<<<(DINP * L1_ + 255) / 256, 256, 0, stream>>>(w1, w1t);

  // ---- encoder layer 1: Linear -> BN -> ReLU ----
  wmma_gemm<1, 0, false><<<dim3(1, MT), 128, 0, stream>>>(
      xs, DINP, w1t, L1_, b1, hp1, L1_, B_, L1_, DINP);
  bn_stats<<<L1_, 256, 0, stream>>>(hp1, L1_, gamma1, beta1, sc1, sh1);
  bn_apply_relu<<<(B_ * L1_ + 255) / 256, 256, 0, stream>>>(hp1, L1_, sc1, sh1);

  // ---- encoder layer 2: Linear -> BN -> ReLU ----
  wmma_gemm<1, 0, true><<<dim3(1, MT), 128, 0, stream>>>(
      hp1, L1_, w2, L1_, b2, hp2, L2_, B_, L2_, L1_);
  bn_stats<<<L2_, 256, 0, stream>>>(hp2, L2_, gamma2, beta2, sc2, sh2);
  bn_apply_relu<<<(B_ * L2_ + 255) / 256, 256, 0, stream>>>(hp2, L2_, sc2, sh2);

  // ---- gating MLP: ELU, ELU, softmax ----
  wmma_gemm<1, 1, true><<<dim3(1, MT), 128, 0, stream>>>(
      hp2, L2_, gw1, L2_, gb1, g1, GH_, B_, GH_, L2_);
  wmma_gemm<1, 1, true><<<dim3(1, MT), 128, 0, stream>>>(
      g1, GH_, gw2, GH_, gb2, g2, GH_, B_, GH_, GH_);
  wmma_gemm<1, 0, true><<<dim3(1, MT), 128, 0, stream>>>(
      g2, GH_, gw3, GH_, gb3, bc, E_, B_, E_, GH_);
  softmax8<<<(B_ + 255) / 256, 256, 0, stream>>>(bc);

  // ---- generator layer 1: K = E*DIN = 824 ----
  build_a1<<<B_, 128, 0, stream>>>(xs, bc, eb1, A1, RB1);
  wmma_gemm_gen<1><<<dim3(H_ / 64, MB), 128, 0, stream>>>(
      A1, KG1, ew1, H_, RB1, H_, h1, H_, B_, H_, KG1);

  // ---- generator layer 2: K = E*H = 2048 ----
  build_a2<<<B_, 128, 0, stream>>>(h1, bc, eb2, H_, A2, RB2);
  wmma_gemm_gen<1><<<dim3(H_ / 64, MB), 128, 0, stream>>>(
      A2, KG2, ew2, H_, RB2, H_, h2, H_, B_, H_, KG2);

  // ---- generator layer 3 (output): K = 2048, N = 51 ----
  build_a2<<<B_, 128, 0, stream>>>(h2, bc, eb3, DOUT_, A2, RB3);
  wmma_gemm_gen<0><<<dim3(1, MB), 128, 0, stream>>>(
      A2, KG2, ew3, DOUT_, RB3, DOUT_, out, DOUT_, B_, DOUT_, KG2);
}